// gconv_19301583028599
// MI455X (gfx1250) — compile-verified
//
#include <hip/hip_runtime.h>

typedef float v2f __attribute__((ext_vector_type(2)));
typedef float v4f __attribute__((ext_vector_type(4)));
typedef float v8f __attribute__((ext_vector_type(8)));

#define N_NODES 8192
#define EMBED   16
#define BATCH   32
#define HID     64
#define TILE    16
#define WAVES   8
#define NCH2    (N_NODES / 32)     // 256 double-chunks of 32 m
#define XT_BYTES ((size_t)(BATCH * 2) * N_NODES * sizeof(float))  // 2 MB

// ---- stage X as xt[m/16][j][m%16], j = 2*b + i : makes V A-slices contiguous ----
__global__ __launch_bounds__(256)
void xpose_kernel(const float* __restrict__ x, float* __restrict__ xt) {
  const int o    = blockIdx.x * 256 + threadIdx.x;   // [0, 524288)
  const int ml   = o & 15;
  const int j    = (o >> 4) & 63;
  const int mblk = o >> 10;
  const int m    = mblk * 16 + ml;
  const int b    = j >> 1;
  const int i    = j & 1;
  xt[o] = x[((size_t)b * N_NODES + m) * 2 + i];
}

// P^T (D-layout) -> B-operand slices; one shfl serves both half-waves.
__device__ __forceinline__ void make_pb(const float* p, int h, v2f* pb) {
  const float d0 = __shfl_xor(h ? p[0] : p[2], 16, 32);
  const float d1 = __shfl_xor(h ? p[1] : p[3], 16, 32);
  const float d2 = __shfl_xor(h ? p[4] : p[6], 16, 32);
  const float d3 = __shfl_xor(h ? p[5] : p[7], 16, 32);
  if (h == 0) {
    pb[0].x = p[0]; pb[0].y = p[1];  pb[1].x = p[4]; pb[1].y = p[5];
    pb[2].x = d0;   pb[2].y = d1;    pb[3].x = d2;   pb[3].y = d3;
  } else {
    pb[0].x = d0;   pb[0].y = d1;    pb[1].x = d2;   pb[1].y = d3;
    pb[2].x = p[2]; pb[2].y = p[3];  pb[3].x = p[6]; pb[3].y = p[7];
  }
}

template <bool XT>
__global__ __launch_bounds__(WAVES * 32)
void agcrn_gconv_kernel(const float* __restrict__ x,      // [B, N, 2]
                        const float* __restrict__ adj,    // [N, 16]
                        const float* __restrict__ wpool,  // [16, 2, 2, 64]
                        const float* __restrict__ bpool,  // [16, 64]
                        const float* __restrict__ xt,     // staged V (or null)
                        float* __restrict__ out)          // [B, N, 64]
{
  __shared__ float lds_O[WAVES][4][8][32];
  __shared__ float lds_m[WAVES][32];
  __shared__ float lds_l[WAVES][32];
  __shared__ float lds_xg[16][64];           // normalized (ADJ @ X)^T tile
  __shared__ float lds_W[16][2][2][64];
  __shared__ float lds_bias[16][64];

  const int tid  = threadIdx.x;
  const int lane = tid & 31;
  const int wave = tid >> 5;
  const int r16  = lane & 15;
  const int h    = lane >> 4;
  const int n0   = blockIdx.x * TILE;

  const v8f vzero = {0.f, 0.f, 0.f, 0.f, 0.f, 0.f, 0.f, 0.f};

  // Q as B-operand slices: B[k][n] = E[n0+n][k]
  v2f qb[4];
#pragma unroll
  for (int kc = 0; kc < 4; ++kc)
    qb[kc] = *(const v2f*)(adj + (n0 + r16) * EMBED + 4 * kc + 2 * h);

  float m_run = 0.0f, l_run = 0.0f;   // relu() => true row max >= 0
  v8f O[4];
#pragma unroll
  for (int t = 0; t < 4; ++t) O[t] = vzero;

  for (int ch = wave; ch < NCH2; ch += WAVES) {
    const int m0  = ch * 32;
    const int mb0 = ch * 2;           // 16-row block index into xt

    // two K-tiles as A-operand slices
    v2f ka0[4], ka1[4];
#pragma unroll
    for (int kc = 0; kc < 4; ++kc) {
      ka0[kc] = *(const v2f*)(adj + (m0 + r16) * EMBED + 4 * kc + 2 * h);
      ka1[kc] = *(const v2f*)(adj + (m0 + 16 + r16) * EMBED + 4 * kc + 2 * h);
    }

    // S^T[m'][n] for both 16-m halves (chained K=4 fp32 WMMAs)
    v8f s0 = vzero, s1 = vzero;
#pragma unroll
    for (int kc = 0; kc < 4; ++kc)
      s0 = __builtin_amdgcn_wmma_f32_16x16x4_f32(false, ka0[kc], false, qb[kc],
                                                 (short)0, s0, false, false);
#pragma unroll
    for (int kc = 0; kc < 4; ++kc)
      s1 = __builtin_amdgcn_wmma_f32_16x16x4_f32(false, ka1[kc], false, qb[kc],
                                                 (short)0, s1, false, false);

    // relu + one online-softmax update per 32 columns
    float cmax = 0.0f;
#pragma unroll
    for (int r = 0; r < 8; ++r) { s0[r] = fmaxf(s0[r], 0.0f); cmax = fmaxf(cmax, s0[r]); }
#pragma unroll
    for (int r = 0; r < 8; ++r) { s1[r] = fmaxf(s1[r], 0.0f); cmax = fmaxf(cmax, s1[r]); }
    cmax = fmaxf(cmax, __shfl_xor(cmax, 16, 32));
    const float m_new = fmaxf(m_run, cmax);
    const float scale = __expf(m_run - m_new);
    m_run = m_new;

    float p0[8], p1[8], psum = 0.0f;
#pragma unroll
    for (int r = 0; r < 8; ++r) { p0[r] = __expf(s0[r] - m_new); psum += p0[r]; }
#pragma unroll
    for (int r = 0; r < 8; ++r) { p1[r] = __expf(s1[r] - m_new); psum += p1[r]; }
    psum += __shfl_xor(psum, 16, 32);
    l_run = l_run * scale + psum;

#pragma unroll
    for (int t = 0; t < 4; ++t)
#pragma unroll
      for (int r = 0; r < 8; ++r) O[t][r] *= scale;

    v2f pb0[4], pb1[4];
    make_pb(p0, h, pb0);
    make_pb(p1, h, pb1);

    // O^T[j][n] += V^T x P^T for both m-halves
#pragma unroll
    for (int t = 0; t < 4; ++t) {
      const int j = t * 16 + r16;      // j = 2*b + i
#pragma unroll
      for (int mh = 0; mh < 2; ++mh) {
#pragma unroll
        for (int kc = 0; kc < 4; ++kc) {
          v2f av;
          if (XT) {
            av = *(const v2f*)(xt + ((size_t)(mb0 + mh) * 64 + j) * 16 + 4 * kc + 2 * h);
          } else {
            const int b = j >> 1;
            const int i = j & 1;
            const v4f xv = *(const v4f*)(
                x + ((size_t)b * N_NODES + (m0 + mh * 16 + 4 * kc + 2 * h)) * 2);
            av.x = i ? xv[1] : xv[0];
            av.y = i ? xv[3] : xv[2];
          }
          O[t] = __builtin_amdgcn_wmma_f32_16x16x4_f32(
              false, av, false, mh ? pb1[kc] : pb0[kc], (short)0, O[t], false, false);
        }
      }
    }
  }

  // ---------------- cross-wave combine of online-softmax partials ----------------
#pragma unroll
  for (int t = 0; t < 4; ++t)
#pragma unroll
    for (int r = 0; r < 8; ++r) lds_O[wave][t][r][lane] = O[t][r];
  lds_m[wave][lane] = m_run;
  lds_l[wave][lane] = l_run;
  __syncthreads();

  if (wave < 4) {                      // wave w merges j-tile t = w
    const int t = wave;
    float M = lds_m[0][lane];
#pragma unroll
    for (int w = 1; w < WAVES; ++w) M = fmaxf(M, lds_m[w][lane]);
    float fac[WAVES];
    float L = 0.0f;
#pragma unroll
    for (int w = 0; w < WAVES; ++w) {
      fac[w] = __expf(lds_m[w][lane] - M);
      L += fac[w] * lds_l[w][lane];
    }
    const float invL = 1.0f / L;
#pragma unroll
    for (int r = 0; r < 8; ++r) {
      float acc = 0.0f;
#pragma unroll
      for (int w = 0; w < WAVES; ++w) acc += fac[w] * lds_O[w][t][r][lane];
      const int j = t * 16 + r + 8 * h;
      lds_xg[r16][j] = acc * invL;
    }
  }

  // ---------------- node-adaptive weights / bias ----------------
  for (int idx = tid; idx < 16 * 2 * 2 * HID; idx += WAVES * 32) {
    const int o = idx & 63;
    const int i = (idx >> 6) & 1;
    const int k = (idx >> 7) & 1;
    const int n = idx >> 8;
    float acc = 0.0f;
#pragma unroll
    for (int d = 0; d < EMBED; ++d)
      acc += adj[(n0 + n) * EMBED + d] * wpool[((d * 2 + k) * 2 + i) * HID + o];
    lds_W[n][k][i][o] = acc;
  }
  for (int idx = tid; idx < 16 * HID; idx += WAVES * 32) {
    const int o = idx & 63;
    const int n = idx >> 6;
    float acc = 0.0f;
#pragma unroll
    for (int d = 0; d < EMBED; ++d)
      acc += adj[(n0 + n) * EMBED + d] * bpool[d * HID + o];
    lds_bias[n][o] = acc;
  }
  __syncthreads();

  // ---------------- fused output ----------------
  for (int pp = tid; pp < 16 * HID; pp += WAVES * 32) {
    const int o = pp & 63;
    const int n = pp >> 6;
    const float w00 = lds_W[n][0][0][o], w01 = lds_W[n][0][1][o];
    const float w10 = lds_W[n][1][0][o], w11 = lds_W[n][1][1][o];
    const float bz  = lds_bias[n][o];
    const size_t row = (size_t)(n0 + n);
#pragma unroll 4
    for (int b = 0; b < BATCH; ++b) {
      const v2f xv = *(const v2f*)(x + ((size_t)b * N_NODES + row) * 2);
      const float g0 = lds_xg[n][2 * b];
      const float g1 = lds_xg[n][2 * b + 1];
      out[((size_t)b * N_NODES + row) * HID + o] =
          bz + xv.x * w00 + xv.y * w01 + g0 * w10 + g1 * w11;
    }
  }
}

extern "C" void kernel_launch(void* const* d_in, const int* in_sizes, int n_in,
                              void* d_out, int out_size, void* d_ws, size_t ws_size,
                              hipStream_t stream) {
  (void)in_sizes; (void)n_in; (void)out_size;
  const float* x     = (const float*)d_in[0];
  const float* adj   = (const float*)d_in[1];
  const float* wpool = (const float*)d_in[2];
  const float* bpool = (const float*)d_in[3];
  float* out = (float*)d_out;

  dim3 grid(N_NODES / TILE);   // 512 n-tiles
  dim3 block(WAVES * 32);      // 8 wave32 per tile

  if (ws_size >= XT_BYTES && d_ws != nullptr) {
    float* xt = (float*)d_ws;
    hipLaunchKernelGGL(xpose_kernel, dim3((BATCH * 2 * N_NODES) / 256), dim3(256),
                       0, stream, x, xt);
    hipLaunchKernelGGL((agcrn_gconv_kernel<true>), grid, block, 0, stream,
                       x, adj, wpool, bpool, (const float*)xt, out);
  } else {
    hipLaunchKernelGGL((agcrn_gconv_kernel<false>), grid, block, 0, stream,
                       x, adj, wpool, bpool, (const float*)nullptr, out);
  }
}